// GNN_420906795476
// MI455X (gfx1250) — compile-verified
//
#include <hip/hip_runtime.h>

// Problem constants (shapes fixed by the reference)
#define HH 128      // hidden/features
#define GG 32       // global features
#define LL 3        // layers
#define BN_EPS 1e-5f

typedef float v2f __attribute__((ext_vector_type(2)));
typedef float v8f __attribute__((ext_vector_type(8)));

__device__ __forceinline__ void atomAddF(float* p, float v) {
    // lowers to global_atomic_add_f32 (relaxed, device scope)
    __hip_atomic_fetch_add(p, v, __ATOMIC_RELAXED, __HIP_MEMORY_SCOPE_AGENT);
}

// ---------------------------------------------------------------- utilities
__global__ void zero4_kernel(float4* __restrict__ p, long n4) {
    long i = (long)blockIdx.x * blockDim.x + threadIdx.x;
    long stride = (long)gridDim.x * blockDim.x;
    float4 z = make_float4(0.f, 0.f, 0.f, 0.f);
    for (; i < n4; i += stride) p[i] = z;
}

__global__ void count_kernel(const int* __restrict__ ids,
                             unsigned* __restrict__ cnt, int n) {
    int i = blockIdx.x * blockDim.x + threadIdx.x;
    if (i < n) atomicAdd(&cnt[ids[i]], 1u);
}

__global__ void invcnt_kernel(const unsigned* __restrict__ cnt,
                              float* __restrict__ inv, int n) {
    int i = blockIdx.x * blockDim.x + threadIdx.x;
    if (i < n) {
        unsigned c = cnt[i];
        inv[i] = c ? 1.f / (float)c : 0.f;
    }
}

// fold BN running stats: y = x*scale + shift
__global__ void bnprep_kernel(const float* __restrict__ g, const float* __restrict__ b,
                              const float* __restrict__ m, const float* __restrict__ v,
                              float* __restrict__ scale, float* __restrict__ shift, int n) {
    int i = blockIdx.x * blockDim.x + threadIdx.x;
    if (i < n) {
        float s = g[i] * rsqrtf(v[i] + BN_EPS);
        scale[i] = s;
        shift[i] = b[i] - m[i] * s;
    }
}

// ------------------------------------------------- edge scatter (sum to agg)
// one wave per edge: 32 lanes x float4 = full 128-float row, coalesced
__global__ __launch_bounds__(256) void scatter_kernel(const float* __restrict__ h,
                                                      const int* __restrict__ src,
                                                      const int* __restrict__ dst,
                                                      float* __restrict__ agg, int E) {
    long t = (long)blockIdx.x * 256 + threadIdx.x;
    int e = (int)(t >> 5);
    if (e >= E) return;
    int fg = ((int)t & 31) << 2;
    const float4 val = *(const float4*)(h + (long)src[e] * HH + fg);
    float* p = agg + (long)dst[e] * HH + fg;
    atomAddF(p + 0, val.x);
    atomAddF(p + 1, val.y);
    atomAddF(p + 2, val.z);
    atomAddF(p + 3, val.w);
}

// ------------------------------------------- pooling scatter (sum over batch)
__global__ __launch_bounds__(256) void pool_kernel(const float* __restrict__ h,
                                                   const int* __restrict__ batch,
                                                   float* __restrict__ pooled, int N) {
    long t = (long)blockIdx.x * 256 + threadIdx.x;
    int n = (int)(t >> 5);
    if (n >= N) return;
    int fg = ((int)t & 31) << 2;
    const float4 val = *(const float4*)(h + (long)n * HH + fg);
    float* p = pooled + (long)batch[n] * HH + fg;
    atomAddF(p + 0, val.x);
    atomAddF(p + 1, val.y);
    atomAddF(p + 2, val.z);
    atomAddF(p + 3, val.w);
}

// ------------------------------------------------------------- fused WMMA GEMM
// hout = relu( ((agg*inv)@wl + hin@wr + bias) * scale + shift )
// Block: 256 threads = 8 waves. Block owns 16 rows; wave w owns cols [16w,16w+16).
// A tiles (scaled agg, hin) staged in LDS (rows padded to 132 floats); result is
// transposed back through LDS for fully-coalesced b128 stores.
__global__ __launch_bounds__(256) void gemm_bn_relu_kernel(
        const float* __restrict__ agg, const float* __restrict__ invc,
        const float* __restrict__ hin,
        const float* __restrict__ wl, const float* __restrict__ wr,
        const float* __restrict__ bias,
        const float* __restrict__ scale, const float* __restrict__ shift,
        float* __restrict__ hout, int N) {
    __shared__ float sA[16 * 132];
    __shared__ float sH[16 * 132];

    const int tid = threadIdx.x;
    const int row0 = blockIdx.x * 16;
    const bool full = (row0 + 16) <= N;   // uniform per block

    // cooperative tile load: 16 x 128 x 2 arrays, agg pre-scaled by 1/deg
    for (int i = tid; i < 16 * HH; i += 256) {
        int r = i >> 7, c = i & (HH - 1);
        int gr = row0 + r;
        float a = 0.f, hv = 0.f;
        if (full || gr < N) {
            long o = (long)gr * HH + c;
            a = agg[o] * invc[gr];
            hv = hin[o];
        }
        sA[r * 132 + c] = a;
        sH[r * 132 + c] = hv;
    }
    __syncthreads();

    const int wave = tid >> 5;   // 0..7 -> col tile
    const int lane = tid & 31;
    const int m = lane & 15;     // row within tile (A), col within tile (B/D)
    const int kg = lane >> 4;    // K-pair group select
    const int col = wave * 16 + m;

    v8f acc = {};
#pragma unroll 4
    for (int kb = 0; kb < HH; kb += 4) {
        const int k0 = kb + 2 * kg;
        // A: lane holds A[m][k0], A[m][k0+1]   (ISA 16x4 f32 layout)
        v2f aA = *(const v2f*)&sA[m * 132 + k0];
        v2f aH = *(const v2f*)&sH[m * 132 + k0];
        // B: lane holds B[k0][col], B[k0+1][col]
        v2f bL, bR;
        bL.x = wl[(long)k0 * HH + col];
        bL.y = wl[(long)(k0 + 1) * HH + col];
        bR.x = wr[(long)k0 * HH + col];
        bR.y = wr[(long)(k0 + 1) * HH + col];
        acc = __builtin_amdgcn_wmma_f32_16x16x4_f32(false, aA, false, bL,
                                                    (short)0, acc, false, false);
        acc = __builtin_amdgcn_wmma_f32_16x16x4_f32(false, aH, false, bR,
                                                    (short)0, acc, false, false);
    }

    // epilogue in registers: +bias, BN, ReLU (lane owns column `col`)
    const float bb = bias[col], sc = scale[col], sh = shift[col];
    float y[8];
#pragma unroll
    for (int v = 0; v < 8; ++v)
        y[v] = fmaxf((acc[v] + bb) * sc + sh, 0.f);

    // transpose tile through LDS (reuse sA) -> coalesced row-major stores
    __syncthreads();            // all waves done reading sA/sH
#pragma unroll
    for (int v = 0; v < 8; ++v)
        sA[(v + 8 * kg) * 132 + col] = y[v];   // D layout: VGPR v -> row v+8*kg
    __syncthreads();

    if (full) {
        // 16 rows x 128 cols = 512 float4 stores, 2 per thread, fully coalesced
#pragma unroll
        for (int i = tid; i < 16 * (HH / 4); i += 256) {
            int r = i >> 5;              // 32 float4 per row
            int c = (i & 31) << 2;
            const float* s = &sA[r * 132 + c];
            float4 val = make_float4(s[0], s[1], s[2], s[3]);
            *(float4*)&hout[(long)(row0 + r) * HH + c] = val;
        }
    } else {
        for (int i = tid; i < 16 * HH; i += 256) {
            int r = i >> 7, c = i & (HH - 1);
            if (row0 + r < N)
                hout[(long)(row0 + r) * HH + c] = sA[r * 132 + c];
        }
    }
}

// ------------------------------------------------------------------ MLP head
// one block (64 threads) per graph: mean-pool finalize, concat g, fc1+relu, fc2
__global__ __launch_bounds__(64) void head_kernel(
        const float* __restrict__ pooled_sum, const unsigned* __restrict__ cntb,
        const float* __restrict__ gfeat,
        const float* __restrict__ fc1w, const float* __restrict__ fc1b,
        const float* __restrict__ fc2w, const float* __restrict__ fc2b,
        float* __restrict__ out) {
    __shared__ float in[HH + GG];
    __shared__ float hid[64];
    const int g = blockIdx.x, t = threadIdx.x;
    const unsigned c = cntb[g];
    const float inv = c ? 1.f / (float)c : 0.f;
    for (int i = t; i < HH; i += 64) in[i] = pooled_sum[(long)g * HH + i] * inv;
    if (t < GG) in[HH + t] = gfeat[(long)g * GG + t];
    __syncthreads();
    float a = fc1b[t];
#pragma unroll 8
    for (int k = 0; k < HH + GG; ++k) a += in[k] * fc1w[k * 64 + t];
    hid[t] = fmaxf(a, 0.f) * fc2w[t];
    __syncthreads();
    if (t < 32) hid[t] += hid[t + 32];
    __syncthreads();
    if (t == 0) {
        float s = 0.f;
        for (int i = 0; i < 32; ++i) s += hid[i];
        out[g] = s + fc2b[0];
    }
}

// -------------------------------------------------------------------- launch
extern "C" void kernel_launch(void* const* d_in, const int* in_sizes, int n_in,
                              void* d_out, int out_size, void* d_ws, size_t ws_size,
                              hipStream_t stream) {
    const float* x     = (const float*)d_in[0];
    const int*   eidx  = (const int*)d_in[1];
    const int*   batch = (const int*)d_in[2];
    const float* gfeat = (const float*)d_in[3];
    const float* wl    = (const float*)d_in[4];
    const float* wr    = (const float*)d_in[5];
    const float* bias  = (const float*)d_in[6];
    const float* gamma = (const float*)d_in[7];
    const float* beta  = (const float*)d_in[8];
    const float* mean  = (const float*)d_in[9];
    const float* var   = (const float*)d_in[10];
    const float* fc1w  = (const float*)d_in[11];
    const float* fc1b  = (const float*)d_in[12];
    const float* fc2w  = (const float*)d_in[13];
    const float* fc2b  = (const float*)d_in[14];

    const int N = in_sizes[0] / HH;
    const int E = in_sizes[1] / 2;
    const int B = in_sizes[3] / GG;
    const int* src = eidx;
    const int* dst = eidx + E;

    // workspace carve-up
    const size_t NH = (size_t)N * HH;
    char* w = (char*)d_ws;
    float*    agg    = (float*)w;                 w += NH * 4;
    float*    hA     = (float*)w;                 w += NH * 4;
    float*    hB     = (float*)w;                 w += NH * 4;
    float*    invc   = (float*)w;                 w += (size_t)N * 4;
    unsigned* cnt    = (unsigned*)w;              w += (size_t)N * 4;      // |
    unsigned* cntb   = (unsigned*)w;              w += (size_t)B * 4;      // | contiguous
    float*    pooled = (float*)w;                 w += (size_t)B * HH * 4; // | zero region
    float*    scale  = (float*)w;                 w += (size_t)LL * HH * 4;
    float*    shift  = (float*)w;                 w += (size_t)LL * HH * 4;

    // zero counts + pooled (contiguous, multiple of 4 words)
    {
        long words = (long)N + B + (long)B * HH;
        zero4_kernel<<<2048, 256, 0, stream>>>((float4*)cnt, words / 4);
    }
    count_kernel<<<(E + 255) / 256, 256, 0, stream>>>(dst, cnt, E);
    count_kernel<<<(N + 255) / 256, 256, 0, stream>>>(batch, cntb, N);
    invcnt_kernel<<<(N + 255) / 256, 256, 0, stream>>>(cnt, invc, N);
    bnprep_kernel<<<(LL * HH + 127) / 128, 128, 0, stream>>>(gamma, beta, mean, var,
                                                             scale, shift, LL * HH);

    const long scatterBlocks = ((long)E * 32 + 255) / 256;
    const int  gemmBlocks    = (N + 15) / 16;
    const float* hin = x;
    float* houts[LL] = {hA, hB, hA};
    for (int l = 0; l < LL; ++l) {
        zero4_kernel<<<4096, 256, 0, stream>>>((float4*)agg, (long)(NH / 4));
        scatter_kernel<<<(int)scatterBlocks, 256, 0, stream>>>(hin, src, dst, agg, E);
        gemm_bn_relu_kernel<<<gemmBlocks, 256, 0, stream>>>(
            agg, invc, hin,
            wl + (size_t)l * HH * HH, wr + (size_t)l * HH * HH,
            bias + (size_t)l * HH, scale + (size_t)l * HH, shift + (size_t)l * HH,
            houts[l], N);
        hin = houts[l];
    }

    const long poolBlocks = ((long)N * 32 + 255) / 256;
    pool_kernel<<<(int)poolBlocks, 256, 0, stream>>>(hin, batch, pooled, N);
    head_kernel<<<B, 64, 0, stream>>>(pooled, cntb, gfeat, fc1w, fc1b, fc2w, fc2b,
                                      (float*)d_out);
}